// IntraAgg_22909355557119
// MI455X (gfx1250) — compile-verified
//
#include <hip/hip_runtime.h>
#include <math.h>

// ---- problem constants (from reference setup_inputs) ----
#define BATCH   8192
#define NNEIGH  64
#define KSAMP   16
#define FEAT    128
#define CATK    256     // 2*FEAT
#define EMB     64

typedef float v2f __attribute__((ext_vector_type(2)));
typedef float v4f __attribute__((ext_vector_type(4)));
typedef float v8f __attribute__((ext_vector_type(8)));

// LDS row strides padded so stride % 64 == 4 (floats): WMMA fragment reads
// (lanes 0-15 at k, lanes 16-31 at k+2, each a float2) touch all 64 banks once.
#define CAT_STRIDE 260
#define WT_STRIDE  260

// ================= Kernel 1: top-16 smallest |score diff| per row ==========
// One wave32 per batch row; 2 candidates per lane; 16 rounds of lexicographic
// (diff, idx) wave-min. Matches jax top_k(-diff) order: ascending diff,
// ties broken by lower neighbor index.
__global__ __launch_bounds__(256) void select_kernel(
    const float* __restrict__ batch_scores,   // [B,2]
    const float* __restrict__ neigh_scores,   // [B,64,2]
    const int*   __restrict__ neighs_ids,     // [B,64]
    int*         __restrict__ sel_ids,        // [B,16]  (workspace)
    float*       __restrict__ samp_out)       // [B,16]  (d_out + B*EMB)
{
    const int lane = threadIdx.x & 31;
    const int wv   = threadIdx.x >> 5;
    const int b    = blockIdx.x * 8 + wv;

    const float cs = batch_scores[b * 2];

    const int i0 = lane;
    const int i1 = lane + 32;
    float d0 = fabsf(cs - neigh_scores[(b * NNEIGH + i0) * 2]);
    float d1 = fabsf(cs - neigh_scores[(b * NNEIGH + i1) * 2]);

    for (int s = 0; s < KSAMP; ++s) {
        float md; int mi;
        if (d0 < d1 || (d0 == d1 && i0 < i1)) { md = d0; mi = i0; }
        else                                  { md = d1; mi = i1; }
        // wave32 lexicographic min reduction
        for (int off = 16; off >= 1; off >>= 1) {
            float od = __shfl_xor(md, off, 32);
            int   oi = __shfl_xor(mi, off, 32);
            if (od < md || (od == md && oi < mi)) { md = od; mi = oi; }
        }
        if (lane == 0) {
            samp_out[b * KSAMP + s] = md;
            sel_ids[b * KSAMP + s]  = neighs_ids[b * NNEIGH + mi];
        }
        // knock out the winner (exactly one lane owns candidate index mi)
        if (mi == i0) d0 = INFINITY;
        if (mi == i1) d1 = INFINITY;
    }
}

// ========= Kernel 2: gather + mean + concat + GEMM(relu) via f32 WMMA ======
// 32 batch rows per workgroup, 256 threads = 8 waves.
//  LDS: cat[32][260]  (self||mean-of-16 features, f32)
//       wt [64][260]  (weight transposed: wt[col][k])
// Each wave computes one 16x16 tile of out = relu(cat @ W): 2 Mtiles x 4 Ntiles,
// K=256 -> 64 chained V_WMMA_F32_16X16X4_F32.
__global__ __launch_bounds__(256) void agg_gemm_kernel(
    const float* __restrict__ features,  // [N_NODES,128]
    const float* __restrict__ weight,    // [256,64] row-major
    const int*   __restrict__ nodes,     // [B]
    const int*   __restrict__ sel_ids,   // [B,16]
    float*       __restrict__ out)       // [B,64]
{
    extern __shared__ float smem[];
    float* cat = smem;                       // 32*260 floats
    float* wt  = smem + 32 * CAT_STRIDE;     // 64*260 floats

    const int t  = threadIdx.x;
    const int b0 = blockIdx.x * 32;

    // ---- stage self features: 32 rows x 128 floats = 1024 float4 ----
    #pragma unroll
    for (int i = 0; i < 4; ++i) {
        int f  = t + 256 * i;        // 0..1023
        int r  = f >> 5;
        int c4 = f & 31;
        v4f v = *reinterpret_cast<const v4f*>(
            features + (size_t)nodes[b0 + r] * FEAT + c4 * 4);
        *reinterpret_cast<v4f*>(&cat[r * CAT_STRIDE + c4 * 4]) = v;
    }

    // ---- gather + mean: 32 rows x 8 threads; each thread owns 16 feats ----
    {
        const int r    = t >> 3;
        const int tsub = t & 7;
        v4f a0 = {0.f,0.f,0.f,0.f}, a1 = {0.f,0.f,0.f,0.f};
        v4f a2 = {0.f,0.f,0.f,0.f}, a3 = {0.f,0.f,0.f,0.f};
        const int* ids = sel_ids + (size_t)(b0 + r) * KSAMP;
        #pragma unroll 4
        for (int s = 0; s < KSAMP; ++s) {
            const float* src = features + (size_t)ids[s] * FEAT + tsub * 16;
            a0 += *reinterpret_cast<const v4f*>(src + 0);
            a1 += *reinterpret_cast<const v4f*>(src + 4);
            a2 += *reinterpret_cast<const v4f*>(src + 8);
            a3 += *reinterpret_cast<const v4f*>(src + 12);
        }
        const float inv = 1.0f / (float)KSAMP;
        float* dst = &cat[r * CAT_STRIDE + FEAT + tsub * 16];
        *reinterpret_cast<v4f*>(dst + 0)  = a0 * inv;
        *reinterpret_cast<v4f*>(dst + 4)  = a1 * inv;
        *reinterpret_cast<v4f*>(dst + 8)  = a2 * inv;
        *reinterpret_cast<v4f*>(dst + 12) = a3 * inv;
    }

    // ---- stage weight transposed: wt[col][k] = W[k][col] ----
    #pragma unroll
    for (int i = 0; i < 16; ++i) {
        int f  = t + 256 * i;       // float4 index, 0..4095
        int k  = f >> 4;            // 16 float4 per 64-wide row
        int c4 = f & 15;
        v4f w = *reinterpret_cast<const v4f*>(weight + k * EMB + c4 * 4);
        wt[(c4 * 4 + 0) * WT_STRIDE + k] = w.x;
        wt[(c4 * 4 + 1) * WT_STRIDE + k] = w.y;
        wt[(c4 * 4 + 2) * WT_STRIDE + k] = w.z;
        wt[(c4 * 4 + 3) * WT_STRIDE + k] = w.w;
    }

    __syncthreads();

    // ---- WMMA: wave -> one 16x16 tile ----
    const int wvid  = t >> 5;
    const int lane  = t & 31;
    const int mtile = wvid >> 2;            // 0..1
    const int ntile = wvid & 3;             // 0..3
    const int L     = lane & 15;
    const int koff  = (lane >> 4) << 1;     // lanes 16-31 hold k+2,k+3

    const float* arow = &cat[(mtile * 16 + L) * CAT_STRIDE + koff];
    const float* brow = &wt [(ntile * 16 + L) * WT_STRIDE  + koff];

    v8f c = {0.f,0.f,0.f,0.f,0.f,0.f,0.f,0.f};
    #pragma unroll 8
    for (int k0 = 0; k0 < CATK; k0 += 4) {
        v2f a = *reinterpret_cast<const v2f*>(arow + k0);
        v2f b = *reinterpret_cast<const v2f*>(brow + k0);
        // 8 args: (neg_a, A, neg_b, B, c_mod, C, reuse_a, reuse_b)
        c = __builtin_amdgcn_wmma_f32_16x16x4_f32(
                false, a, false, b, (short)0, c, false, false);
    }

    // ---- ReLU + store (C/D layout: vgpr v -> M=v (lanes 0-15) / v+8) ----
    const int col   = ntile * 16 + L;
    const int rbase = b0 + mtile * 16 + ((lane >> 4) << 3);
    #pragma unroll
    for (int v = 0; v < 8; ++v) {
        float x = c[v];
        out[(size_t)(rbase + v) * EMB + col] = x > 0.f ? x : 0.f;
    }
}

// ============================ launcher =====================================
extern "C" void kernel_launch(void* const* d_in, const int* in_sizes, int n_in,
                              void* d_out, int out_size, void* d_ws, size_t ws_size,
                              hipStream_t stream) {
    const float* features = (const float*)d_in[0];   // [200000,128]
    const float* weight   = (const float*)d_in[1];   // [256,64]
    const int*   nodes    = (const int*)d_in[2];     // [8192]
    const float* bscores  = (const float*)d_in[3];   // [8192,2]
    const float* nscores  = (const float*)d_in[4];   // [8192,64,2]
    const int*   nids     = (const int*)d_in[5];     // [8192,64]
    // d_in[6] = num_sample scalar; fixed at 16 in the reference setup.
    (void)in_sizes; (void)n_in; (void)out_size; (void)ws_size;

    float* out_feats = (float*)d_out;                      // [8192,64]
    float* out_samp  = out_feats + (size_t)BATCH * EMB;    // [8192,16]
    int*   sel       = (int*)d_ws;                         // [8192,16]

    select_kernel<<<BATCH / 8, 256, 0, stream>>>(bscores, nscores, nids, sel, out_samp);

    size_t smem = (size_t)(32 * CAT_STRIDE + 64 * WT_STRIDE) * sizeof(float); // ~98 KB
    agg_gemm_kernel<<<BATCH / 32, 256, smem, stream>>>(features, weight, nodes, sel, out_feats);
}